// MGIoU2DMinus_59760174957246
// MI455X (gfx1250) — compile-verified
//
#include <hip/hip_runtime.h>
#include <hip/hip_bf16.h>

// CDNA5 / gfx1250, wave32. The pairwise MGIoU matrix is symmetric
// (g[i,j]==g[j,i]: same 4-interval GIoU set with roles swapped), so one wave
// computes one UPPER-TRIANGULAR 16x16 tile via twelve v_wmma_f32_16x16x4_f32
// ops (K=2 padded to 4) and writes both the tile and its transpose (through a
// 16x17 LDS tile so both store patterns stay coalesced). This halves WMMA and
// VALU work for a compute-bound kernel; store bytes (67 MB) are unchanged.

typedef __attribute__((ext_vector_type(2))) float v2f;
typedef __attribute__((ext_vector_type(8))) float v8f;

#define WMMA4(A, B, C)                                                      \
    __builtin_amdgcn_wmma_f32_16x16x4_f32(false, (A), false, (B), (short)0, \
                                          (C), false, false)

// Per-box precomputed record, 16 floats (64 B, 16B-aligned fields):
//  [0] cx  [1] cy   [2] e1x [3] e1y  [4] e2x [5] e2y
//  [6] u0x [7] u0y  [8] u1x [9] u1y  [10,11] pad
//  [12] s0 (ctr.u0) [13] rad0 (w*w/2) [14] s1 (ctr.u1) [15] rad1 (h*h/2)
__global__ __launch_bounds__(256) void mgiou_precompute(
    const float* __restrict__ boxes, float* __restrict__ tab, int n) {
  int b = blockIdx.x * blockDim.x + threadIdx.x;
  if (b >= n) return;
  float cx = boxes[5 * b + 0];
  float cy = boxes[5 * b + 1];
  float w  = boxes[5 * b + 2];
  float h  = boxes[5 * b + 3];
  float ang = boxes[5 * b + 4];
  float s, c;
  sincosf(ang, &s, &c);
  // corners = ctr + a*E1 + b*E2, a,b in {-1,1}
  float e1x = 0.5f * w * c, e1y = -0.5f * w * s;  // (w/2)*(c,-s)
  float e2x = 0.5f * h * s, e2y = 0.5f * h * c;   // (h/2)*(s, c)
  float u0x = 2.f * e1x, u0y = 2.f * e1y;  // axis0 = corners[1]-corners[0]
  float u1x = 2.f * e2x, u1y = 2.f * e2y;  // axis1 = corners[3]-corners[0]
  float s0 = cx * u0x + cy * u0y;
  float s1 = cx * u1x + cy * u1y;
  float rad0 = 0.5f * w * w;  // |E1.u0| (E2.u0 == 0)
  float rad1 = 0.5f * h * h;
  float4* T4 = (float4*)(tab + (size_t)b * 16);
  T4[0] = make_float4(cx, cy, e1x, e1y);
  T4[1] = make_float4(e2x, e2y, u0x, u0y);
  T4[2] = make_float4(u1x, u1y, 0.f, 0.f);
  T4[3] = make_float4(s0, rad0, s1, rad1);
}

// Symmetric 1-D GIoU on intervals (p-r, p+r) and (s2-r2, s2+r2).
// inter/union - (hull-union)/hull = (inter*hull - (hull-union)*union)
//                                   / (union*hull)   -> one fast rcp.
__device__ __forceinline__ float giou1d(float p, float r, float s2, float r2) {
  float min1 = p - r, max1 = p + r;
  float min2 = s2 - r2, max2 = s2 + r2;
  float inter = fmaxf(0.f, fminf(max1, max2) - fmaxf(min1, min2));
  float uni = 2.f * r + 2.f * r2 - inter;
  float hull = fmaxf(max1, max2) - fminf(min1, min2);
  float num = inter * hull - (hull - uni) * uni;
  return num * __builtin_amdgcn_rcpf(uni * hull);
}

template <int N>  // N > 0: compile-time row stride; N == 0: runtime stride
__global__ __launch_bounds__(32) void mgiou_pairs(
    const float* __restrict__ tab, float* __restrict__ out, int n) {
  // Symmetric output: only tiles with it <= jt do work (uniform wave exit,
  // keeps EXEC all-ones for the WMMAs in surviving waves).
  if (blockIdx.y > blockIdx.x) return;

  __shared__ float tr[16][17];  // transposed tile staging (pad -> few conflicts)

  const int lane = threadIdx.x;  // 0..31, wave32
  const int lcol = lane & 15;    // column j within tile
  const int half = lane >> 4;    // 0: rows 0-7, 1: rows 8-15
  const int jt = blockIdx.x << 4;
  const int it = blockIdx.y << 4;
  const bool lo = (half == 0);  // lanes 16-31 supply K=2,3 -> zeros

  const float* Ti = tab + (size_t)(it + lcol) * 16;
  const float* Tj = tab + (size_t)(jt + lcol) * 16;

  // A operands: rows = i boxes; B operands: columns = j boxes.
  // 16x4 f32 A layout: lanes 0-15 hold K=0,1 in vgpr0,1; lanes 16-31 K=2,3.
  // Only K=0,1 are used (2-D dots); upper-half lanes carry zeros.
#define LOAD2(dst, Tp, off)     \
  {                             \
    float x0 = (Tp)[(off)];     \
    float x1 = (Tp)[(off) + 1]; \
    dst[0] = lo ? x0 : 0.f;     \
    dst[1] = lo ? x1 : 0.f;     \
  }
  v2f Actr, Ae1, Ae2, Au0, Au1;
  v2f Bctr, Be1, Be2, Bu0, Bu1;
  LOAD2(Actr, Ti, 0) LOAD2(Ae1, Ti, 2) LOAD2(Ae2, Ti, 4)
  LOAD2(Au0, Ti, 6)  LOAD2(Au1, Ti, 8)
  LOAD2(Bctr, Tj, 0) LOAD2(Be1, Tj, 2) LOAD2(Be2, Tj, 4)
  LOAD2(Bu0, Tj, 6)  LOAD2(Bu1, Tj, 8)
#undef LOAD2

  v8f z = {0.f, 0.f, 0.f, 0.f, 0.f, 0.f, 0.f, 0.f};
  // Direction B (box i corners onto box j axes): P = ctr_i.u_ja, R = E_i.u_ja
  v8f P0  = WMMA4(Actr, Bu0, z);
  v8f P1  = WMMA4(Actr, Bu1, z);
  v8f R10 = WMMA4(Ae1, Bu0, z);
  v8f R11 = WMMA4(Ae1, Bu1, z);
  v8f R20 = WMMA4(Ae2, Bu0, z);
  v8f R21 = WMMA4(Ae2, Bu1, z);
  // Direction A (box j corners onto box i axes): Q = u_ia.ctr_j, S = u_ia.E_j
  v8f Q0  = WMMA4(Au0, Bctr, z);
  v8f Q1  = WMMA4(Au1, Bctr, z);
  v8f S10 = WMMA4(Au0, Be1, z);
  v8f S11 = WMMA4(Au1, Be1, z);
  v8f S20 = WMMA4(Au0, Be2, z);
  v8f S21 = WMMA4(Au1, Be2, z);

  // Self-projection scalars of this lane's j box: (s0, rad0, s1, rad1)
  float4 js = *(const float4*)(Tj + 12);
  const int j = jt + lcol;
  const int irow0 = it + (half << 3);  // first row owned by this lane half

  // One base pointer; slot loads/stores become base + immediate offset.
  const float* Tis = tab + (size_t)irow0 * 16 + 12;
  const size_t stride = (N > 0) ? (size_t)N : (size_t)n;
  float* orow = out + (size_t)irow0 * stride + j;

#pragma unroll
  for (int s = 0; s < 8; ++s) {
    float4 is = *(const float4*)(Tis + s * 16);

    float rB0 = fabsf(R10[s]) + fabsf(R20[s]);
    float rB1 = fabsf(R11[s]) + fabsf(R21[s]);
    float rA0 = fabsf(S10[s]) + fabsf(S20[s]);
    float rA1 = fabsf(S11[s]) + fabsf(S21[s]);

    float g0 = giou1d(P0[s], rB0, js.x, js.y);  // j's axis 0
    float g1 = giou1d(P1[s], rB1, js.z, js.w);  // j's axis 1
    float g2 = giou1d(Q0[s], rA0, is.x, is.y);  // i's axis 0
    float g3 = giou1d(Q1[s], rA1, is.z, is.w);  // i's axis 1

    float g = fminf(fminf(g0, g1), fminf(g2, g3));
    g = fmaxf(g, 0.f);
    if (irow0 + s == j) g = 0.f;  // zero the diagonal
    orow[s * stride] = g;         // direct block, coalesced
    tr[lcol][(half << 3) + s] = g;  // stage for transposed block
  }

  __syncthreads();  // cross-lane LDS visibility before transpose read

  // Mirror block: out[j, i] = g[i, j], rows jt.., cols it.. (coalesced).
  float* orowT = out + (size_t)(jt + (half << 3)) * stride + it + lcol;
#pragma unroll
  for (int s = 0; s < 8; ++s) {
    orowT[s * stride] = tr[(half << 3) + s][lcol];
  }
}

extern "C" void kernel_launch(void* const* d_in, const int* in_sizes, int n_in,
                              void* d_out, int out_size, void* d_ws,
                              size_t ws_size, hipStream_t stream) {
  const float* boxes = (const float*)d_in[0];
  float* out = (float*)d_out;
  float* tab = (float*)d_ws;      // needs n*16*4 bytes (256 KB for n=4096)
  const int n = in_sizes[0] / 5;  // 4096

  mgiou_precompute<<<(n + 255) / 256, 256, 0, stream>>>(boxes, tab, n);

  const int nt = (n + 15) / 16;
  dim3 grid(nt, nt);  // one wave32 per tile; lower-triangle waves exit early
  if (n == 4096) {
    mgiou_pairs<4096><<<grid, dim3(32, 1, 1), 0, stream>>>(tab, out, n);
  } else {
    mgiou_pairs<0><<<grid, dim3(32, 1, 1), 0, stream>>>(tab, out, n);
  }
}